// SNN_60567628808223
// MI455X (gfx1250) — compile-verified
//
#include <hip/hip_runtime.h>

typedef __attribute__((ext_vector_type(2))) float v2f;
typedef __attribute__((ext_vector_type(8))) float v8f;

namespace {
constexpr float kBeta = 0.9f;
constexpr float kThr  = 1.0f;
constexpr int kT = 50;
constexpr int kB = 16384;
constexpr int kD = 784;
constexpr int kH = 128;
constexpr int kO = 10;

constexpr int kWaves     = 4;
constexpr int kRowsPerWG = kWaves * 16;  // 64 batch rows per workgroup
constexpr int kKC        = 16;           // K-chunk (784 = 49 * 16)
constexpr int kNChunks   = kD / kKC;     // 49
constexpr int kDS        = 20;           // LDS row stride (floats): 80B -> 16B-aligned b128, low bank conflict
constexpr int kSS        = 130;          // LDS row stride (floats) for spikes / W2^T: conflict-free b64
}  // namespace

__device__ __forceinline__ v8f splat8(float x) {
  v8f r = {x, x, x, x, x, x, x, x};
  return r;
}

// CDNA5 async copy: 16B per lane, global -> LDS, tracked by ASYNCcnt.
__device__ __forceinline__ void async_copy_b128(unsigned lds_off, unsigned voff,
                                                const float* sbase) {
  asm volatile("global_load_async_to_lds_b128 %0, %1, %2"
               :: "v"(lds_off), "v"(voff), "s"(sbase)
               : "memory");
}

__device__ __forceinline__ void wait_async_zero() {
  asm volatile("s_wait_asynccnt 0x0" ::: "memory");
}

__global__ __launch_bounds__(kWaves * 32)
void snn_fused_kernel(const float* __restrict__ data,
                      const float* __restrict__ W1,
                      const float* __restrict__ b1,
                      const float* __restrict__ W2,
                      const float* __restrict__ b2,
                      float* __restrict__ out) {
  __shared__ float sData[2][kRowsPerWG * kDS];  // double-buffered 64x16 data chunk
  __shared__ float sW1[2][kH * kDS];            // double-buffered 128x16 W1 chunk
  __shared__ float sW2t[16 * kSS];              // W2 padded to 16 output cols
  __shared__ float sSpk[kWaves * 16 * kSS];     // per-wave 16x128 spike tile

  const int tid  = threadIdx.x;
  const int wave = tid >> 5;
  const int lane = tid & 31;
  const int lm   = lane & 15;   // M-row (A/C) or N-col (B/C) within a 16-tile
  const int kh   = lane >> 4;   // K-half selector for A/B fragments
  const int row0 = blockIdx.x * kRowsPerWG;

  // Stage zero-padded W2 (10x128 -> 16x128) once; read-only afterwards.
  for (int s = tid; s < 16 * kH; s += blockDim.x) {
    const int o = s >> 7, k = s & (kH - 1);
    sW2t[o * kSS + k] = (o < kO) ? W2[o * kH + k] : 0.0f;
  }

  // Per-thread async staging slots (2 data b128s + 4 W1 b128s per chunk).
  unsigned d_voff[2], d_lds[2];
#pragma unroll
  for (int j = 0; j < 2; ++j) {
    const int s = tid + j * (kWaves * 32);
    const int r = s >> 2, q = s & 3;
    d_voff[j] = (unsigned)(((size_t)(row0 + r) * kD + q * 4) * 4);
    d_lds[j]  = (unsigned)((r * kDS + q * 4) * 4);
  }
  unsigned w_voff[4], w_lds[4];
#pragma unroll
  for (int j = 0; j < 4; ++j) {
    const int s = tid + j * (kWaves * 32);
    const int n = s >> 2, q = s & 3;
    w_voff[j] = (unsigned)((n * kD + q * 4) * 4);
    w_lds[j]  = (unsigned)((n * kDS + q * 4) * 4);
  }
  const unsigned sData_base = (unsigned)(size_t)(void*)&sData[0][0];
  const unsigned sW1_base   = (unsigned)(size_t)(void*)&sW1[0][0];
  constexpr unsigned kDataBufBytes = kRowsPerWG * kDS * 4;
  constexpr unsigned kW1BufBytes   = kH * kDS * 4;

  auto issue_chunk = [&](int chunk, int buf) {
    const unsigned kcB = (unsigned)(chunk * kKC * 4);
    const unsigned db = sData_base + (unsigned)buf * kDataBufBytes;
    const unsigned wb = sW1_base + (unsigned)buf * kW1BufBytes;
#pragma unroll
    for (int j = 0; j < 2; ++j) async_copy_b128(db + d_lds[j], d_voff[j] + kcB, data);
#pragma unroll
    for (int j = 0; j < 4; ++j) async_copy_b128(wb + w_lds[j], w_voff[j] + kcB, W1);
  };

  // cur1 accumulators: 8 N-tiles of 16x16 f32 C fragments; seed with b1.
  v8f cur1[8];
#pragma unroll
  for (int n = 0; n < 8; ++n) cur1[n] = splat8(b1[n * 16 + lm]);

  // ---------------- Phase A: cur1 = data @ W1^T + b1 (double-buffered async) ----------------
  issue_chunk(0, 0);
  for (int ic = 0; ic < kNChunks; ++ic) {
    wait_async_zero();   // my stage(ic) complete
    __syncthreads();     // all waves staged ic; all waves done computing ic-1
    if (ic + 1 < kNChunks) issue_chunk(ic + 1, (ic + 1) & 1);  // overlaps compute below

    const float* bd = &sData[ic & 1][0];
    const float* bw = &sW1[ic & 1][0];
#pragma unroll
    for (int k4 = 0; k4 < kKC; k4 += 4) {
      const int kk = k4 + 2 * kh;  // even -> 8B-aligned ds_load_b64
      // A 16x4: lane holds row lm, consecutive K pair (K-half by kh).
      v2f a = *(const v2f*)&bd[(wave * 16 + lm) * kDS + kk];
#pragma unroll
      for (int n = 0; n < 8; ++n) {
        // B 4x16: lane holds col lm; B[k][col] = W1[col][k].
        v2f b = *(const v2f*)&bw[(n * 16 + lm) * kDS + kk];
        cur1[n] = __builtin_amdgcn_wmma_f32_16x16x4_f32(
            false, a, false, b, (short)0, cur1[n], false, false);
      }
    }
  }

  // ---------------- Phase B: 50-step LIF recurrence, in-register ----------------
  const float b2v = (lm < kO) ? b2[lm] : 0.0f;
  v8f mem1[8];
#pragma unroll
  for (int n = 0; n < 8; ++n) mem1[n] = splat8(0.0f);
  v8f mem2 = splat8(0.0f);
  float* myspk = &sSpk[wave * 16 * kSS];  // per-wave private region

  for (int t = 0; t < kT; ++t) {
    // Layer-1 LIF update; spikes -> per-wave LDS tile in A-fragment layout.
    // No barrier: only this wave touches myspk; LDS is in-order within a wave.
#pragma unroll
    for (int n = 0; n < 8; ++n) {
#pragma unroll
      for (int v = 0; v < 8; ++v) {
        float m = mem1[n][v];
        const float rst = (m > kThr) ? kThr : 0.0f;
        m = kBeta * m + cur1[n][v] - rst;
        mem1[n][v] = m;
        myspk[(v + 8 * kh) * kSS + n * 16 + lm] = (m > kThr) ? 1.0f : 0.0f;
      }
    }

    // Layer-2 GEMM: cur2(16x16, cols 0..9 valid) = spk1(16x128) @ W2^T + b2.
    v8f c2 = splat8(b2v);
#pragma unroll
    for (int k4 = 0; k4 < kH; k4 += 4) {
      const int kk = k4 + 2 * kh;
      v2f a = *(const v2f*)&myspk[lm * kSS + kk];
      v2f b = *(const v2f*)&sW2t[lm * kSS + kk];
      c2 = __builtin_amdgcn_wmma_f32_16x16x4_f32(
          false, a, false, b, (short)0, c2, false, false);
    }

    // Layer-2 LIF update; spk2 straight to global (the only HBM stores).
#pragma unroll
    for (int v = 0; v < 8; ++v) {
      float m = mem2[v];
      const float rst = (m > kThr) ? kThr : 0.0f;
      m = kBeta * m + c2[v] - rst;
      mem2[v] = m;
      if (lm < kO) {
        const int row = row0 + wave * 16 + v + 8 * kh;
        out[((size_t)t * kB + row) * kO + lm] = (m > kThr) ? 1.0f : 0.0f;
      }
    }
  }
}

extern "C" void kernel_launch(void* const* d_in, const int* in_sizes, int n_in,
                              void* d_out, int out_size, void* d_ws, size_t ws_size,
                              hipStream_t stream) {
  const float* data = (const float*)d_in[0];
  const float* W1   = (const float*)d_in[1];
  const float* b1   = (const float*)d_in[2];
  const float* W2   = (const float*)d_in[3];
  const float* b2   = (const float*)d_in[4];
  float* out = (float*)d_out;

  dim3 grid(kB / kRowsPerWG);  // 256 workgroups
  dim3 block(kWaves * 32);     // 128 threads = 4 waves (wave32)
  snn_fused_kernel<<<grid, block, 0, stream>>>(data, W1, b1, W2, b2, out);
}